// KNNClassifier_41979010351652
// MI455X (gfx1250) — compile-verified
//
#include <hip/hip_runtime.h>

typedef float v2f __attribute__((ext_vector_type(2)));
typedef float v8f __attribute__((ext_vector_type(8)));

#define TILE_M 16
#define K1_BLOCKS 480
#define K1_THREADS 256
#define WAVES_PER_BLOCK (K1_THREADS / 32)
#define TOPK 15
#define K2_THREADS 256
#define NCAND (K1_BLOCKS * TOPK)   // 7200 candidates -> 57.6KB LDS in finisher
#define NUM_CLASSES 3
#define MAX_D 512

__device__ __forceinline__ void topk_insert(unsigned long long (&best)[TOPK],
                                            unsigned long long key) {
    if (key < best[TOPK - 1]) {
        best[TOPK - 1] = key;
#pragma unroll
        for (int i = TOPK - 1; i > 0; --i) {
            if (best[i] < best[i - 1]) {
                unsigned long long t = best[i];
                best[i] = best[i - 1];
                best[i - 1] = t;
            }
        }
    }
}

// Wave-per-16-row-tile distance kernel. Cross term via V_WMMA_F32_16X16X4_F32,
// |t|^2 in VALU on the same loaded data. Query staged to LDS via the CDNA5
// async global->LDS path; fused hierarchical top-K.
__global__ __launch_bounds__(K1_THREADS)
void knn_dist_topk_kernel(const float* __restrict__ x,
                          const float* __restrict__ train,
                          int n, int d,
                          unsigned long long* __restrict__ blockCand) {
    __shared__ __align__(16) float xs[MAX_D];   // query staged in LDS

    const int tid  = threadIdx.x;
    const int lane = tid & 31;
    const int waveInBlock = tid >> 5;
    const int gwave  = blockIdx.x * WAVES_PER_BLOCK + waveInBlock;
    const int nwaves = gridDim.x * WAVES_PER_BLOCK;

    // Stage x into LDS. D==128 -> exactly 32 lanes x 16B: a single
    // global_load_async_to_lds_b128 from wave 0 (ASYNCcnt path).
    if (d == 128) {
        if (waveInBlock == 0) {
            unsigned int      ldsa = (unsigned int)(uintptr_t)xs + (unsigned int)(lane * 16);
            unsigned long long ga  = (unsigned long long)(uintptr_t)x +
                                     (unsigned long long)(lane * 16);
            asm volatile("global_load_async_to_lds_b128 %0, %1, off"
                         :: "v"(ldsa), "v"(ga) : "memory");
            asm volatile("s_wait_asynccnt 0" ::: "memory");
        }
    } else {
        for (int i = tid; i < d && i < MAX_D; i += K1_THREADS) xs[i] = x[i];
    }
    __syncthreads();

    // |x|^2 from LDS (once per thread)
    float xsq = 0.f;
    for (int i = 0; i < d; ++i) { float xv = xs[i]; xsq += xv * xv; }

    const int half = lane >> 4;    // A/B K-half: lanes 0-15 -> K{0,1}, 16-31 -> K{2,3}
    const int mrow = lane & 15;    // A row owned by this lane

    unsigned long long best[TOPK];
#pragma unroll
    for (int i = 0; i < TOPK; ++i) best[i] = ~0ull;

    const int ntiles = (n + TILE_M - 1) / TILE_M;
    for (int t = gwave; t < ntiles; t += nwaves) {
        const int rowBase = t * TILE_M;
        const int row     = rowBase + mrow;
        const bool valid  = (row < n);
        const long rowSafe = valid ? (long)row : (long)(n - 1);
        const float* rp = train + rowSafe * (long)d;

        // prefetch next tile's row for this lane (global_prefetch_b8)
        if (t + nwaves < ntiles) {
            const float* np = train + ((long)(t + nwaves) * TILE_M + mrow) * (long)d;
            __builtin_prefetch(np, 0, 3);
            __builtin_prefetch(np + 64, 0, 3);
        }

        v8f acc = {0.f, 0.f, 0.f, 0.f, 0.f, 0.f, 0.f, 0.f};
        float ssq = 0.f;
#pragma unroll 8
        for (int k0 = 0; k0 < d; k0 += 4) {
            const int c = k0 + 2 * half;
            v2f a = *(const v2f*)(rp + c);         // 8B aligned global (HBM stream)
            if (!valid) { a[0] = 0.f; a[1] = 0.f; }    // select, not branch: EXEC stays full
            v2f b = *(const v2f*)(xs + c);         // ds_load_b64 broadcast within half
            acc = __builtin_amdgcn_wmma_f32_16x16x4_f32(
                false, a, false, b, (short)0, acc, false, false);
            ssq += a[0] * a[0] + a[1] * a[1];
        }
        // lane l and lane l^16 together covered all 128 cols of row (l&15)
        float tsq = ssq + __shfl_xor(ssq, 16, 32);

        // C layout: VGPR v holds rows v (lanes 0-15) and v+8 (lanes 16-31);
        // all 16 N-columns identical (B broadcast) -> any lane of the half works.
        float myCross = 0.f, myTsq = 0.f;
#pragma unroll
        for (int m = 0; m < 16; ++m) {
            float cm = __shfl(acc[m & 7], (m < 8) ? 0 : 16, 32);
            float tm = __shfl(tsq, m, 32);
            if (lane == m) { myCross = cm; myTsq = tm; }
        }
        if (lane < 16 && row < n) {
            float d2 = myTsq - 2.0f * myCross + xsq;
            d2 = fmaxf(d2, 0.f);
            unsigned long long key =
                ((unsigned long long)__float_as_uint(d2) << 32) | (unsigned int)row;
            topk_insert(best, key);
        }
    }

    // Block-level top-K: 15 rounds of parallel min over 256x15 candidates.
    __shared__ unsigned long long cand[TOPK * K1_THREADS];  // 30.7KB
    __shared__ unsigned long long redv[K1_THREADS];
    __shared__ int redi[K1_THREADS];
#pragma unroll
    for (int i = 0; i < TOPK; ++i) cand[i * K1_THREADS + tid] = best[i];
    __syncthreads();
    for (int r = 0; r < TOPK; ++r) {
        unsigned long long mv = ~0ull; int mi = tid;
#pragma unroll
        for (int i = 0; i < TOPK; ++i) {
            int idx = i * K1_THREADS + tid;
            unsigned long long v = cand[idx];
            if (v < mv) { mv = v; mi = idx; }
        }
        redv[tid] = mv; redi[tid] = mi;
        __syncthreads();
        for (int s = K1_THREADS / 2; s > 0; s >>= 1) {
            if (tid < s) {
                if (redv[tid + s] < redv[tid]) {
                    redv[tid] = redv[tid + s];
                    redi[tid] = redi[tid + s];
                }
            }
            __syncthreads();
        }
        if (tid == 0) {
            blockCand[blockIdx.x * TOPK + r] = redv[0];
            cand[redi[0]] = ~0ull;
        }
        __syncthreads();
    }
}

// Single-block finisher: global top-15 over 7200 candidates, label vote, argmax.
__global__ __launch_bounds__(K2_THREADS)
void knn_final_kernel(const unsigned long long* __restrict__ blockCand,
                      const int* __restrict__ labels,
                      int n, float* __restrict__ out) {
    __shared__ unsigned long long cand[NCAND];     // 57.6KB
    __shared__ unsigned long long redv[K2_THREADS];
    __shared__ int redi[K2_THREADS];
    __shared__ unsigned long long top[TOPK];
    const int tid = threadIdx.x;

    for (int i = tid; i < NCAND; i += K2_THREADS) cand[i] = blockCand[i];
    __syncthreads();

    for (int r = 0; r < TOPK; ++r) {
        unsigned long long mv = ~0ull; int mi = 0;
        for (int i = tid; i < NCAND; i += K2_THREADS) {
            unsigned long long v = cand[i];
            if (v < mv) { mv = v; mi = i; }
        }
        redv[tid] = mv; redi[tid] = mi;
        __syncthreads();
        for (int s = K2_THREADS / 2; s > 0; s >>= 1) {
            if (tid < s) {
                if (redv[tid + s] < redv[tid]) {
                    redv[tid] = redv[tid + s];
                    redi[tid] = redi[tid + s];
                }
            }
            __syncthreads();
        }
        if (tid == 0) { top[r] = redv[0]; cand[redi[0]] = ~0ull; }
        __syncthreads();
    }

    if (tid == 0) {
        int counts[NUM_CLASSES] = {0, 0, 0};
        for (int r = 0; r < TOPK; ++r) {
            unsigned long long k = top[r];
            if (k == ~0ull) continue;
            unsigned int idx = (unsigned int)(k & 0xFFFFFFFFu);
            if (idx < (unsigned int)n) {
                int lab = labels[idx];
                if (lab >= 0 && lab < NUM_CLASSES) counts[lab]++;
            }
        }
        int bestc = 0, bestv = -1;
        for (int c = 0; c < NUM_CLASSES; ++c) {
            if (counts[c] > bestv) { bestv = counts[c]; bestc = c; }  // first-max tie rule
        }
        out[0] = (float)bestc;
    }
}

extern "C" void kernel_launch(void* const* d_in, const int* in_sizes, int n_in,
                              void* d_out, int out_size, void* d_ws, size_t ws_size,
                              hipStream_t stream) {
    const float* x      = (const float*)d_in[0];   // [D]
    const float* train  = (const float*)d_in[1];   // [N, D]
    const int*   labels = (const int*)d_in[2];     // [N]
    const int d = in_sizes[0];                     // 128
    const int n = in_sizes[2];                     // 2,000,000

    unsigned long long* blockCand = (unsigned long long*)d_ws;  // NCAND u64 = 57.6KB

    knn_dist_topk_kernel<<<K1_BLOCKS, K1_THREADS, 0, stream>>>(x, train, n, d, blockCand);
    knn_final_kernel<<<1, K2_THREADS, 0, stream>>>(blockCand, labels, n, (float*)d_out);
}